// CorrelationLoss_20650202759362
// MI455X (gfx1250) — compile-verified
//
#include <hip/hip_runtime.h>
#include <math.h>

// Problem constants from the reference (setup_inputs is fixed).
#define NS 10          // student views
#define NT 2           // teacher views
#define BB 128         // batch
#define S_INVT 10.0f   // 1/STUDENT_TEMP
#define T_INVT 25.0f   // 1/TEACHER_TEMP
#define SPLIT 4        // D-range split per batch element for occupancy
#define PAYLOAD 44     // [0..1] tm, [2..3] tS, [4..23] tD[t*10+s], [24..33] sm, [34..43] ss
#define CORR_WAVES 128 // waves cooperating on the 16x16xD Gram

typedef __attribute__((ext_vector_type(2))) float v2f;
typedef __attribute__((ext_vector_type(8))) float v8f;

// ---------------------------------------------------------------------------
// Kernel 1: fused single-pass per-(b, d-slice):
//   - teacher online softmax stats (m, S) with rescaled dot accumulators
//     D[t][s] = sum_d exp(z_t[d]-m) * x_s[d]   (so dot = D/S, exact)
//   - student online logsumexp (m, s) of x/Ts
// One block per (b, slice). Deterministic tree reductions in LDS.
// Every HBM byte of student/teacher is touched exactly once.
// ---------------------------------------------------------------------------
__global__ void fused_dot_lse(const float* __restrict__ stu,
                              const float* __restrict__ tea,
                              const float* __restrict__ cen,
                              float* __restrict__ pay, int D) {
  const int b = blockIdx.x;
  const int part = blockIdx.y;
  const int tid = threadIdx.x;
  const int chunk = D / SPLIT;
  const int dBeg = part * chunk;
  const int dEnd = dBeg + chunk;

  float tm[NT], tS[NT], tD[NT][NS];
  float sm[NS], ss[NS];
#pragma unroll
  for (int t = 0; t < NT; ++t) {
    tm[t] = -INFINITY; tS[t] = 0.f;
#pragma unroll
    for (int s = 0; s < NS; ++s) tD[t][s] = 0.f;
  }
#pragma unroll
  for (int s = 0; s < NS; ++s) { sm[s] = -INFINITY; ss[s] = 0.f; }

  const float* trow[NT];
#pragma unroll
  for (int t = 0; t < NT; ++t) trow[t] = tea + ((size_t)t * BB + b) * (size_t)D;
  const float* srow[NS];
#pragma unroll
  for (int s = 0; s < NS; ++s) srow[s] = stu + ((size_t)s * BB + b) * (size_t)D;

  for (int d = dBeg + tid * 4; d < dEnd; d += blockDim.x * 4) {
    float4 c4 = *(const float4*)(cen + d);
    float4 x4[NS];
#pragma unroll
    for (int s = 0; s < NS; ++s) x4[s] = *(const float4*)(srow[s] + d);

#pragma unroll
    for (int t = 0; t < NT; ++t) {
      float4 y4 = *(const float4*)(trow[t] + d);
      float z0 = (y4.x - c4.x) * T_INVT, z1 = (y4.y - c4.y) * T_INVT;
      float z2 = (y4.z - c4.z) * T_INVT, z3 = (y4.w - c4.w) * T_INVT;
      float zc = fmaxf(fmaxf(z0, z1), fmaxf(z2, z3));
      if (zc > tm[t]) {  // rescale running sums (mathematically exact)
        float f = expf(tm[t] - zc);
        tS[t] *= f;
#pragma unroll
        for (int s = 0; s < NS; ++s) tD[t][s] *= f;
        tm[t] = zc;
      }
      float w0 = expf(z0 - tm[t]), w1 = expf(z1 - tm[t]);
      float w2 = expf(z2 - tm[t]), w3 = expf(z3 - tm[t]);
      tS[t] += (w0 + w1) + (w2 + w3);
#pragma unroll
      for (int s = 0; s < NS; ++s)
        tD[t][s] += w0 * x4[s].x + w1 * x4[s].y + w2 * x4[s].z + w3 * x4[s].w;
    }

#pragma unroll
    for (int s = 0; s < NS; ++s) {
      float z0 = x4[s].x * S_INVT, z1 = x4[s].y * S_INVT;
      float z2 = x4[s].z * S_INVT, z3 = x4[s].w * S_INVT;
      float zc = fmaxf(fmaxf(z0, z1), fmaxf(z2, z3));
      if (zc > sm[s]) { ss[s] *= expf(sm[s] - zc); sm[s] = zc; }
      ss[s] += expf(z0 - sm[s]) + expf(z1 - sm[s]) + expf(z2 - sm[s]) + expf(z3 - sm[s]);
    }
  }

  __shared__ float red[256 * PAYLOAD];
  float* my = red + tid * PAYLOAD;
  my[0] = tm[0]; my[1] = tm[1]; my[2] = tS[0]; my[3] = tS[1];
#pragma unroll
  for (int t = 0; t < NT; ++t)
#pragma unroll
    for (int s = 0; s < NS; ++s) my[4 + t * NS + s] = tD[t][s];
#pragma unroll
  for (int s = 0; s < NS; ++s) { my[24 + s] = sm[s]; my[34 + s] = ss[s]; }
  __syncthreads();

  for (int off = 128; off >= 1; off >>= 1) {
    if (tid < off) {
      float* a = red + tid * PAYLOAD;
      float* o = red + (tid + off) * PAYLOAD;
#pragma unroll
      for (int t = 0; t < NT; ++t) {
        float m1 = a[t], m2 = o[t], m = fmaxf(m1, m2);
        float f1 = expf(m1 - m), f2 = expf(m2 - m);
        a[t] = m;
        a[2 + t] = a[2 + t] * f1 + o[2 + t] * f2;
#pragma unroll
        for (int s = 0; s < NS; ++s) {
          int i = 4 + t * NS + s;
          a[i] = a[i] * f1 + o[i] * f2;
        }
      }
#pragma unroll
      for (int s = 0; s < NS; ++s) {
        float m1 = a[24 + s], m2 = o[24 + s], m = fmaxf(m1, m2);
        a[34 + s] = a[34 + s] * expf(m1 - m) + o[34 + s] * expf(m2 - m);
        a[24 + s] = m;
      }
    }
    __syncthreads();
  }
  if (tid < PAYLOAD)
    pay[((size_t)b * SPLIT + part) * PAYLOAD + tid] = red[tid];
}

// ---------------------------------------------------------------------------
// Kernel 2: Gram matrix G = e0 * e0^T (10x10 padded to 16x16, K = D)
// via V_WMMA_F32_16X16X4_F32.
//   - A (16x4) and B (4x16) VGPR layouts make the per-lane operand pair
//     identical for a symmetric Gram: lane L supplies e0[L&15] at its K pair.
//   - A dot product is K-permutation invariant, so one b128 load per lane
//     (low lanes at d0, high lanes at d0+4) feeds TWO WMMAs:
//       wmma0 <- (x,y): K-set {d0,d0+1,d0+4,d0+5}
//       wmma1 <- (z,w): K-set {d0+2,d0+3,d0+6,d0+7}
//   - Padding lanes (r>=10) load a clamped valid row and multiply by a 0 mask
//     (no EXEC save/restore in the hot loop).
//   - Next iteration's float4 is prefetched before the WMMAs; two independent
//     accumulators break the WMMA RAW chain.
// 128 waves each own a D/128 K-slice; partial 16x16 tiles go to workspace.
// ---------------------------------------------------------------------------
__global__ void corr_gram(const float* __restrict__ stu,
                          float* __restrict__ part, int D) {
  const int lane = threadIdx.x & 31;
  const int wave = threadIdx.x >> 5;
  const int gw = blockIdx.x * (blockDim.x >> 5) + wave;
  const int span = D / CORR_WAVES;      // 512 for D=65536
  const int iters = span / 8;           // 8 d-values per iteration
  const int r = lane & 15;
  const int hi = lane >> 4;             // 0: low half, 1: high half
  const float mask = (r < NS) ? 1.0f : 0.0f;
  // Clamped row pointer: padding lanes read row 0 (valid) and mask to zero.
  const float* base = stu + (size_t)(r < NS ? r : 0) * (size_t)D
                      + (size_t)gw * span + hi * 4;

  v8f c0 = {}, c1 = {};
  float4 cur = *(const float4*)base;
  for (int i = 0; i < iters; ++i) {
    float4 nxt;
    if (i + 1 < iters) {                // uniform branch (scalar)
      nxt = *(const float4*)(base + (size_t)(i + 1) * 8);
    } else {
      nxt = make_float4(0.f, 0.f, 0.f, 0.f);
    }
    v2f a0; a0.x = cur.x * mask; a0.y = cur.y * mask;
    v2f a1; a1.x = cur.z * mask; a1.y = cur.w * mask;
    c0 = __builtin_amdgcn_wmma_f32_16x16x4_f32(false, a0, false, a0,
                                               (short)0, c0, false, false);
    c1 = __builtin_amdgcn_wmma_f32_16x16x4_f32(false, a1, false, a1,
                                               (short)0, c1, false, false);
    cur = nxt;
  }

  const int n = lane & 15;
  const int mbase = hi * 8;
  float* out = part + (size_t)gw * 256;
#pragma unroll
  for (int v = 0; v < 8; ++v) out[(mbase + v) * 16 + n] = c0[v] + c1[v];
}

// ---------------------------------------------------------------------------
// Kernel 3: finalize — reduce Gram partials, cosine-hinge corr loss,
// combine per-b payload slices, assemble dino loss. Single block, fixed
// reduction order => deterministic.
// ---------------------------------------------------------------------------
__global__ void finalize(const float* __restrict__ part,
                         const float* __restrict__ pay,
                         float* __restrict__ out) {
  __shared__ float G[256];
  __shared__ float red[256];
  __shared__ float corr_s;
  const int tid = threadIdx.x;

  float g = 0.f;
  for (int w = 0; w < CORR_WAVES; ++w) g += part[w * 256 + tid];
  G[tid] = g;
  __syncthreads();

  if (tid == 0) {
    float nrm[NS];
#pragma unroll
    for (int i = 0; i < NS; ++i) nrm[i] = fmaxf(sqrtf(G[i * 16 + i]), 1e-12f);
    float cs = 0.f;
    for (int i = 0; i < NS; ++i)
      for (int j = i + 1; j < NS; ++j) {
        float sim = G[i * 16 + j] / (nrm[i] * nrm[j]);
        cs += fmaxf(sim - 0.3f, 0.f);  // threshold = 1 - MARGIN
      }
    corr_s = cs / (float)(NS * (NS - 1) / 2);
  }

  float contrib = 0.f;
  if (tid < BB) {
    const int b = tid;
    float acc[PAYLOAD];
    const float* p0 = pay + (size_t)b * SPLIT * PAYLOAD;
#pragma unroll
    for (int i = 0; i < PAYLOAD; ++i) acc[i] = p0[i];
    for (int k = 1; k < SPLIT; ++k) {
      const float* pk = p0 + k * PAYLOAD;
#pragma unroll
      for (int t = 0; t < NT; ++t) {
        float m1 = acc[t], m2 = pk[t], m = fmaxf(m1, m2);
        float f1 = expf(m1 - m), f2 = expf(m2 - m);
        acc[t] = m;
        acc[2 + t] = acc[2 + t] * f1 + pk[2 + t] * f2;
#pragma unroll
        for (int s = 0; s < NS; ++s) {
          int i = 4 + t * NS + s;
          acc[i] = acc[i] * f1 + pk[i] * f2;
        }
      }
#pragma unroll
      for (int s = 0; s < NS; ++s) {
        float m1 = acc[24 + s], m2 = pk[24 + s], m = fmaxf(m1, m2);
        acc[34 + s] = acc[34 + s] * expf(m1 - m) + pk[34 + s] * expf(m2 - m);
        acc[24 + s] = m;
      }
    }
    float logZs[NS];
#pragma unroll
    for (int s = 0; s < NS; ++s) logZs[s] = acc[24 + s] + logf(acc[34 + s]);
#pragma unroll
    for (int t = 0; t < NT; ++t)
#pragma unroll
      for (int s = 0; s < NS; ++s) {
        if (s == t) continue;  // skip diagonal terms
        float dotN = acc[4 + t * NS + s] / acc[2 + t];  // sum_d t_p * x
        contrib += S_INVT * dotN - logZs[s];            // sum_d t_p * s_logp
      }
  }
  red[tid] = contrib;
  __syncthreads();
  for (int off = 128; off >= 1; off >>= 1) {
    if (tid < off) red[tid] += red[tid + off];
    __syncthreads();
  }
  if (tid == 0) {
    float dino = -red[0] / (float)(BB * (NT * NS - NT));  // 18 off-diag terms
    out[0] = dino + 5.0f * corr_s;  // CORR_WEIGHT = 5
  }
}

// ---------------------------------------------------------------------------
extern "C" void kernel_launch(void* const* d_in, const int* in_sizes, int n_in,
                              void* d_out, int out_size, void* d_ws, size_t ws_size,
                              hipStream_t stream) {
  const float* stu = (const float*)d_in[0];  // (10,128,D)
  const float* tea = (const float*)d_in[1];  // (2,128,D)
  const float* cen = (const float*)d_in[2];  // (D,)
  const int D = in_sizes[2];                 // 65536

  float* ws = (float*)d_ws;
  float* corrPart = ws;                       // CORR_WAVES*256 = 32768 floats
  float* pay = ws + (size_t)CORR_WAVES * 256; // BB*SPLIT*PAYLOAD = 22528 floats

  dim3 g1(BB, SPLIT);
  fused_dot_lse<<<g1, 256, 0, stream>>>(stu, tea, cen, pay, D);
  corr_gram<<<CORR_WAVES / 4, 128, 0, stream>>>(stu, corrPart, D);
  finalize<<<1, 256, 0, stream>>>(corrPart, pay, (float*)d_out);
}